// GCN_adaboost_35871566856597
// MI455X (gfx1250) — compile-verified
//
#include <hip/hip_runtime.h>
#include <hip/hip_bf16.h>

// ---------------------------------------------------------------------------
// CDNA5 / gfx1250 implementation of the GCN-adaboost reference.
// Dense GEMMs use V_WMMA_F32_16X16X4_F32 (full fp32 precision — the problem
// is SpMM-memory-bound, ~6.6 GB scatter/gather vs ~30 GFLOP dense, so fp32
// matrix ops are free). SpMM uses global_atomic_add_f32 scatter.
// GEMM pipeline per K-chunk:
//   1) issue next chunk's global loads (regs),
//   2) batch-load all 32 LDS fragment pairs for this chunk,
//   3) run 16 WMMAs back-to-back on the accumulator chain.
// ---------------------------------------------------------------------------

typedef float v2f __attribute__((ext_vector_type(2)));
typedef float v8f __attribute__((ext_vector_type(8)));

#define GEMM_BM 64   // rows per block (4 waves x 16)
#define GEMM_KT 64   // K chunk staged in LDS (power of two, multiple of 4)

// C[N x M] = A[N x K] @ B[K x M]; M % 16 == 0, K % 4 == 0; N guarded.
__global__ __launch_bounds__(128) void gemm_wmma_f32(
    const float* __restrict__ A, const float* __restrict__ B,
    float* __restrict__ C, int N, int K, int M)
{
    __shared__ float a_s[GEMM_BM][GEMM_KT + 1];   // +1 pad: kill bank conflicts
    __shared__ float b_s[GEMM_KT][16];

    const int tid  = threadIdx.x;
    const int wave = tid >> 5;
    const int lane = tid & 31;
    const int half = lane >> 4;   // which 16-lane half
    const int l16  = lane & 15;
    const int rowBase = blockIdx.x * GEMM_BM;
    const int colBase = blockIdx.y * 16;
    const int arow = wave * 16 + l16;

    constexpr int AR = GEMM_BM * GEMM_KT / 128;   // 32 regs / thread
    constexpr int BR = GEMM_KT * 16 / 128;        // 8 regs / thread
    constexpr int NS = GEMM_KT / 4;               // 16 WMMA steps / chunk
    float areg[AR], breg[BR];

    v8f acc = {};

    const int nfull = K / GEMM_KT;
    const int ktail = K - nfull * GEMM_KT;

    // global -> registers (compile-time trip counts; rows zero-filled OOB)
    auto gload = [&](int kc) {
#pragma unroll
        for (int i = 0; i < AR; ++i) {
            int t  = tid + i * 128;
            int lr = t >> 6;                 // t / GEMM_KT
            int kk = t & (GEMM_KT - 1);
            int gr = rowBase + lr;
            areg[i] = (gr < N) ? A[gr * K + kc + kk] : 0.0f;
        }
#pragma unroll
        for (int i = 0; i < BR; ++i) {
            int t  = tid + i * 128;
            breg[i] = B[(kc + (t >> 4)) * M + colBase + (t & 15)];
        }
    };
    // registers -> LDS
    auto stage = [&]() {
#pragma unroll
        for (int i = 0; i < AR; ++i) {
            int t = tid + i * 128;
            a_s[t >> 6][t & (GEMM_KT - 1)] = areg[i];
        }
#pragma unroll
        for (int i = 0; i < BR; ++i) {
            int t = tid + i * 128;
            b_s[t >> 4][t & 15] = breg[i];
        }
    };
    // One full chunk: batch-load all fragments, then 16 back-to-back WMMAs.
    // ISA 7.12.2: 32-bit A 16x4 frag: lanes0-15 -> K={k,k+1}, lanes16-31 -> K={k+2,k+3}
    auto compute_full = [&]() {
        v2f af[NS], bf[NS];
#pragma unroll
        for (int s = 0; s < NS; ++s) {
            const int kk = s * 4;
            af[s].x = a_s[arow][kk + 2 * half + 0];
            af[s].y = a_s[arow][kk + 2 * half + 1];
            bf[s].x = b_s[kk + 2 * half + 0][l16];
            bf[s].y = b_s[kk + 2 * half + 1][l16];
        }
#pragma unroll
        for (int s = 0; s < NS; ++s) {
            acc = __builtin_amdgcn_wmma_f32_16x16x4_f32(
                false, af[s], false, bf[s], (short)0, acc, false, false);
        }
    };

    if (nfull > 0) {
        gload(0);
        for (int ch = 0; ch < nfull; ++ch) {
            stage();
            __syncthreads();
            if (ch + 1 < nfull) gload((ch + 1) * GEMM_KT);  // overlap with WMMAs
            compute_full();
            __syncthreads();
        }
    }
    if (ktail > 0) {   // rare tail chunk (only K=500 -> 52)
        const int kc = nfull * GEMM_KT;
        for (int t = tid; t < GEMM_BM * ktail; t += 128) {
            int lr = t / ktail, kk = t - lr * ktail;
            int gr = rowBase + lr;
            a_s[lr][kk] = (gr < N) ? A[gr * K + kc + kk] : 0.0f;
        }
        for (int t = tid; t < ktail * 16; t += 128) {
            b_s[t >> 4][t & 15] = B[(kc + (t >> 4)) * M + colBase + (t & 15)];
        }
        __syncthreads();
        for (int kk = 0; kk < ktail; kk += 4) {
            v2f a, b;
            a.x = a_s[arow][kk + 2 * half + 0];
            a.y = a_s[arow][kk + 2 * half + 1];
            b.x = b_s[kk + 2 * half + 0][l16];
            b.y = b_s[kk + 2 * half + 1][l16];
            acc = __builtin_amdgcn_wmma_f32_16x16x4_f32(
                false, a, false, b, (short)0, acc, false, false);
        }
    }

    // C/D frag layout: VGPR i -> M = i + 8*half, N = l16
    if (rowBase + GEMM_BM <= N) {           // 781 of 782 row-blocks: no guards
        float* Cp = C + (rowBase + wave * 16 + 8 * half) * M + colBase + l16;
#pragma unroll
        for (int i = 0; i < 8; ++i) Cp[i * M] = acc[i];
    } else {
#pragma unroll
        for (int i = 0; i < 8; ++i) {
            int gr = rowBase + wave * 16 + 8 * half + i;
            if (gr < N) C[gr * M + colBase + l16] = acc[i];
        }
    }
}

// out[r[e]][:] += v[e] * S[c[e]][:]; one thread per (edge, float4 quad).
__global__ void spmm_scatter(const int* __restrict__ r, const int* __restrict__ c,
                             const float* __restrict__ v, const float* __restrict__ S,
                             float* __restrict__ out, int total, int eshift, int qmask, int M)
{
    int t = blockIdx.x * blockDim.x + threadIdx.x;
    if (t >= total) return;
    int e = t >> eshift;
    int m = (t & qmask) << 2;
    float w = v[e];
    const float4 s4 = *(const float4*)(S + c[e] * M + m);
    float* o = out + r[e] * M + m;
    unsafeAtomicAdd(o + 0, w * s4.x);   // global_atomic_add_f32
    unsafeAtomicAdd(o + 1, w * s4.y);
    unsafeAtomicAdd(o + 2, w * s4.z);
    unsafeAtomicAdd(o + 3, w * s4.w);
}

__global__ void zero_f32(float* __restrict__ p, int n)
{
    int t = blockIdx.x * blockDim.x + threadIdx.x;
    if (t < n) p[t] = 0.0f;
}

__global__ void bias_act(float* __restrict__ h, const float* __restrict__ b,
                         int total, int mmask, int do_relu)
{
    int t = blockIdx.x * blockDim.x + threadIdx.x;
    if (t >= total) return;
    float x = h[t] + b[t & mmask];
    h[t] = do_relu ? fmaxf(x, 0.0f) : x;
}

// out[n][0..6] = b + act(in[n][:]) @ w  (w is K x 7)
__global__ void head7(const float* __restrict__ in, const float* __restrict__ w,
                      const float* __restrict__ b, float* __restrict__ out,
                      int N, int K, int do_relu)
{
    int n = blockIdx.x * blockDim.x + threadIdx.x;
    if (n >= N) return;
    float acc[7];
#pragma unroll
    for (int j = 0; j < 7; ++j) acc[j] = b[j];
    const float* row = in + n * K;
    for (int k = 0; k < K; ++k) {
        float hv = row[k];
        if (do_relu) hv = fmaxf(hv, 0.0f);
#pragma unroll
        for (int j = 0; j < 7; ++j) acc[j] += hv * w[k * 7 + j];
    }
#pragma unroll
    for (int j = 0; j < 7; ++j) out[n * 7 + j] = acc[j];
}

// sim[n] = sb + concat(x2[n], x3[n]) @ sw   (sw is 128 x 7, no relu)
__global__ void sim7(const float* __restrict__ x2, const float* __restrict__ x3,
                     const float* __restrict__ sw, const float* __restrict__ sb,
                     float* __restrict__ out, int N)
{
    int n = blockIdx.x * blockDim.x + threadIdx.x;
    if (n >= N) return;
    float acc[7];
#pragma unroll
    for (int j = 0; j < 7; ++j) acc[j] = sb[j];
    for (int k = 0; k < 64; ++k) {
        float a = x2[n * 64 + k];
        float b = x3[n * 64 + k];
#pragma unroll
        for (int j = 0; j < 7; ++j)
            acc[j] += a * sw[k * 7 + j] + b * sw[(64 + k) * 7 + j];
    }
#pragma unroll
    for (int j = 0; j < 7; ++j) out[n * 7 + j] = acc[j];
}

// te = exp(-(base * yi)), tg = gate * yi;  s_out[0] += te[tg>=0], s_out[1] += te
// pass2 (use_aprev): base := b1 + aprev[0]*b2
__global__ void alpha_reduce(const float* __restrict__ b1, const float* __restrict__ b2,
                             const float* __restrict__ g,  const float* __restrict__ y,
                             const int* __restrict__ idx,  const float* __restrict__ aprev,
                             float* __restrict__ s_out, int total, int use_aprev)
{
    __shared__ float sh1[256], sh2[256];
    int t = blockIdx.x * blockDim.x + threadIdx.x;
    float te = 0.0f, s1 = 0.0f;
    if (t < total) {
        int i = t / 7, j = t - i * 7;
        int ri = idx[i];
        float yv = y[ri * 7 + j];
        float bv = b1[ri * 7 + j];
        if (use_aprev) bv += aprev[0] * b2[ri * 7 + j];
        te = expf(-(bv * yv));
        float tg = g[ri * 7 + j] * yv;
        s1 = (tg >= 0.0f) ? te : 0.0f;
    }
    sh1[threadIdx.x] = s1;
    sh2[threadIdx.x] = te;
    __syncthreads();
    for (int s = 128; s > 0; s >>= 1) {
        if (threadIdx.x < s) {
            sh1[threadIdx.x] += sh1[threadIdx.x + s];
            sh2[threadIdx.x] += sh2[threadIdx.x + s];
        }
        __syncthreads();
    }
    if (threadIdx.x == 0) {
        unsafeAtomicAdd(&s_out[0], sh1[0]);
        unsafeAtomicAdd(&s_out[1], sh2[0]);
    }
}

// a = 0.5 * log((st - s1) / s1)
__global__ void alpha_fin(const float* __restrict__ s, float* __restrict__ a)
{
    if (threadIdx.x == 0 && blockIdx.x == 0) {
        float s1 = s[0];
        float s2 = s[1] - s1;
        a[0] = 0.5f * logf(s2 / s1);
    }
}

// part2 = x4d + a1*sim + a2*x1d
__global__ void combine(const float* __restrict__ x4d, const float* __restrict__ sim,
                        const float* __restrict__ x1d, const float* __restrict__ a1,
                        const float* __restrict__ a2, float* __restrict__ out, int total)
{
    int t = blockIdx.x * blockDim.x + threadIdx.x;
    if (t >= total) return;
    out[t] = x4d[t] + a1[0] * sim[t] + a2[0] * x1d[t];
}

// ---------------------------------------------------------------------------

extern "C" void kernel_launch(void* const* d_in, const int* in_sizes, int n_in,
                              void* d_out, int out_size, void* d_ws, size_t ws_size,
                              hipStream_t stream)
{
    const int NFEAT = 500;
    const int N  = in_sizes[0] / NFEAT;   // 50000
    const int E  = in_sizes[3];           // 800000
    const int NI = in_sizes[2];           // 5000

    const float* x     = (const float*)d_in[0];
    const float* y     = (const float*)d_in[1];
    const int*   index = (const int*)d_in[2];

    // adjacency i in 1..5 at d_in[3 + 3*(i-1) + {0,1,2}]
    const int*   Ar[6]; const int* Ac[6]; const float* Av[6];
    for (int i = 1; i <= 5; ++i) {
        Ar[i] = (const int*)  d_in[3 + 3 * (i - 1) + 0];
        Ac[i] = (const int*)  d_in[3 + 3 * (i - 1) + 1];
        Av[i] = (const float*)d_in[3 + 3 * (i - 1) + 2];
    }
    // W[0..2]=w1..w3, W[3..5]=w4..w6, W[6..8]=w10..w12 (500x128, 128x64, 64x64)
    const float* W[9]; const float* Bv[9];
    for (int i = 0; i < 9; ++i) {
        W[i]  = (const float*)d_in[18 + 2 * i];
        Bv[i] = (const float*)d_in[19 + 2 * i];
    }
    const float* dW[4]; const float* dB[4];
    for (int i = 0; i < 4; ++i) {
        dW[i] = (const float*)d_in[36 + 2 * i];
        dB[i] = (const float*)d_in[37 + 2 * i];
    }
    const float* swp = (const float*)d_in[44];
    const float* sbp = (const float*)d_in[45];

    // workspace layout (floats)
    float* ws = (float*)d_ws;
    size_t off = 0;
    float* S    = ws + off; off += (size_t)N * 128;  // support buffer
    float* hA   = ws + off; off += (size_t)N * 128;  // h1
    float* hB   = ws + off; off += (size_t)N * 64;   // h2
    float* xA   = ws + off; off += (size_t)N * 64;   // branch output slot A
    float* xB   = ws + off; off += (size_t)N * 64;   // branch output slot B
    float* x1d  = ws + off; off += (size_t)N * 7;
    float* simb = ws + off; off += (size_t)N * 7;
    float* x4d  = ws + off; off += (size_t)N * 7;
    float* sums = ws + off; off += 8;                // [s1a, sta, a1, s1b, stb, a2]

    float* out       = (float*)d_out;
    float* x2d_out   = out;
    float* x3d_out   = out + (size_t)N * 7;
    float* part2_out = out + (size_t)2 * N * 7;

    auto gemm = [&](const float* A, const float* Bw, float* Cm, int K, int M) {
        dim3 grid((N + GEMM_BM - 1) / GEMM_BM, M / 16);
        gemm_wmma_f32<<<grid, dim3(128), 0, stream>>>(A, Bw, Cm, N, K, M);
    };
    auto zero = [&](float* p, int n) {
        zero_f32<<<dim3((n + 255) / 256), dim3(256), 0, stream>>>(p, n);
    };
    auto spmm = [&](int ai, const float* Sm, float* o, int M) {
        int quads  = M >> 2;
        int eshift = (M == 128) ? 5 : 4;     // log2(quads)
        int total  = E * quads;
        spmm_scatter<<<dim3((total + 255) / 256), dim3(256), 0, stream>>>(
            Ar[ai], Ac[ai], Av[ai], Sm, o, total, eshift, quads - 1, M);
    };
    auto biasact = [&](float* h, const float* b, int M, int relu) {
        int total = N * M;
        bias_act<<<dim3((total + 255) / 256), dim3(256), 0, stream>>>(
            h, b, total, M - 1, relu);
    };
    // one GCN branch: xout = gcn3(gcn2(gcn1(x)))  (relu inside, none on last)
    auto branch = [&](int ai, const float* Wa, const float* ba,
                      const float* Wb, const float* bb,
                      const float* Wc, const float* bc, float* xout) {
        gemm(x, Wa, S, NFEAT, 128);
        zero(hA, N * 128);
        spmm(ai, S, hA, 128);
        biasact(hA, ba, 128, 1);

        gemm(hA, Wb, S, 128, 64);
        zero(hB, N * 64);
        spmm(ai, S, hB, 64);
        biasact(hB, bb, 64, 1);

        gemm(hB, Wc, S, 64, 64);
        zero(xout, N * 64);
        spmm(ai, S, xout, 64);
        biasact(xout, bc, 64, 0);
    };

    dim3 gN((N + 255) / 256), b256(256);

    // Branch 1: adj a5, w1..w3 -> xA;  x1d = relu(x1) @ d1w + d1b
    branch(5, W[0], Bv[0], W[1], Bv[1], W[2], Bv[2], xA);
    head7<<<gN, b256, 0, stream>>>(xA, dW[0], dB[0], x1d, N, 64, 1);

    // Branch 2: adj a4, w4..w6 -> xA (kept for sim);  x2d -> out
    branch(4, W[3], Bv[3], W[4], Bv[4], W[5], Bv[5], xA);
    head7<<<gN, b256, 0, stream>>>(xA, dW[1], dB[1], x2d_out, N, 64, 1);

    // Branch 3: adj a3, shares w4..w6 -> xB;  x3d -> out
    branch(3, W[3], Bv[3], W[4], Bv[4], W[5], Bv[5], xB);
    head7<<<gN, b256, 0, stream>>>(xB, dW[2], dB[2], x3d_out, N, 64, 1);

    // sim = concat(x2, x3) @ sw + sb
    sim7<<<gN, b256, 0, stream>>>(xA, xB, swp, sbp, simb, N);

    // Branch 4: adj a1, w10..w12 -> xA;  x4d
    branch(1, W[6], Bv[6], W[7], Bv[7], W[8], Bv[8], xA);
    head7<<<gN, b256, 0, stream>>>(xA, dW[3], dB[3], x4d, N, 64, 1);

    // alpha chain
    zero(sums, 8);
    int at = NI * 7;
    dim3 gA((at + 255) / 256);
    alpha_reduce<<<gA, b256, 0, stream>>>(x4d, simb, simb, y, index,
                                          sums + 2, sums + 0, at, 0);
    alpha_fin<<<dim3(1), dim3(32), 0, stream>>>(sums + 0, sums + 2);   // a1
    alpha_reduce<<<gA, b256, 0, stream>>>(x4d, simb, x1d, y, index,
                                          sums + 2, sums + 3, at, 1);
    alpha_fin<<<dim3(1), dim3(32), 0, stream>>>(sums + 3, sums + 5);   // a2

    int ct = N * 7;
    combine<<<dim3((ct + 255) / 256), b256, 0, stream>>>(
        x4d, simb, x1d, sums + 2, sums + 5, part2_out, ct);
}